// GradientExtractor_3487513444492
// MI455X (gfx1250) — compile-verified
//
#include <hip/hip_runtime.h>

typedef __attribute__((ext_vector_type(2))) float v2f;
typedef __attribute__((ext_vector_type(4))) float v4f;
typedef __attribute__((ext_vector_type(8))) float v8f;

#define IMG_W 512
#define IMG_H 512
#define PLANES 96            // 32 batch * 3 channels
#define WPB    8             // waves per block
#define HSTR   36            // H row stride in floats: b128-aligned, 2-way max bank conflict
#define HS     (20 * HSTR)   // per-sigma H plane size

// g[idx] for idx in [0,4] (symmetric taps), else 0 — cndmask chain, no scratch
__device__ __forceinline__ float gsel(int idx, float g0, float g1, float g2) {
    float r = 0.0f;
    r = (idx == 0 || idx == 4) ? g0 : r;
    r = (idx == 1 || idx == 3) ? g1 : r;
    r = (idx == 2) ? g2 : r;
    return r;
}

__global__ __launch_bounds__(256) void dog_wmma_kernel(const float* __restrict__ in,
                                                       float* __restrict__ out) {
    // A-matrix constants for V_WMMA_F32_16X16X4_F32: [sigma][chunk][lane] -> (VGPR0,VGPR1)
    __shared__ v2f   atab[320];             // 2560 B, shared by all waves
    __shared__ float lds_H[WPB][2 * HS];    // row-pass result [sigma][r][x], x = 0..31

    const int tid  = threadIdx.x;
    const int lane = tid & 31;
    const int wave = tid >> 5;
    const int gw   = blockIdx.x * WPB + wave;   // strip id: 16 rows x 32 cols
    const int plane = gw >> 9;                  // / (32 ty * 16 tx-strips)
    const int rem   = gw & 511;
    const int y0    = (rem >> 4) * 16;
    const int x0    = (rem & 15) * 32;

    // ---- normalized separable Gaussian weights (sigmas compile-time) ----
    const float w1a = __expf(-2.0f),  w1b = __expf(-0.5f);
    const float s1  = 2.0f * w1a + 2.0f * w1b + 1.0f;
    const float g1_0 = w1a / s1, g1_1 = w1b / s1, g1_2 = 1.0f / s1;
    const float w2a = __expf(-0.5f),  w2b = __expf(-0.125f);
    const float s2  = 2.0f * w2a + 2.0f * w2b + 1.0f;
    const float g2_0 = w2a / s2, g2_1 = w2b / s2, g2_2 = 1.0f / s2;

    // ---- one-time per-block: A-matrix constant table, A[m][k] = g[k-m], sigma2 negated ----
    for (int e = tid; e < 320; e += 256) {
        int l    = e & 31;
        int c    = (e >> 5) % 5;
        int s    = (e >= 160) ? 1 : 0;
        int M    = l & 15;
        int k0   = 4 * c + ((l < 16) ? 0 : 2);
        float G0 = s ? g2_0 : g1_0, G1 = s ? g2_1 : g1_1, G2 = s ? g2_2 : g1_2;
        float sg = s ? -1.0f : 1.0f;
        v2f a;
        a.x = sg * gsel(k0     - M, G0, G1, G2);
        a.y = sg * gsel(k0 + 1 - M, G0, G1, G2);
        atab[e] = a;
    }

    // ---- row pass, row-per-lane: lanes 0..19 own the 20 patch rows, registers only ----
    const size_t plane_base = (size_t)plane * (IMG_H * IMG_W);
    float* Hb = lds_H[wave];
    if (lane < 20) {
        const int  rg  = y0 - 2 + lane;
        const bool rok = (unsigned)rg < IMG_H;
        const float* rowp = in + plane_base + (size_t)rg * IMG_W;
        float row[40];                                  // cols x0-4 .. x0+35
#pragma unroll
        for (int j = 0; j < 10; ++j) {
            int cg = x0 - 4 + 4 * j;                    // multiple of 4 -> whole-vec4 bounds
            v4f ch = {0.0f, 0.0f, 0.0f, 0.0f};
            if (rok && (unsigned)cg < IMG_W) ch = *(const v4f*)(rowp + cg);
            row[4*j+0] = ch.x; row[4*j+1] = ch.y; row[4*j+2] = ch.z; row[4*j+3] = ch.w;
        }
#pragma unroll
        for (int q = 0; q < 8; ++q) {                   // 4 x-positions per b128 store
            v4f h1, h2;
#pragma unroll
            for (int e2 = 0; e2 < 4; ++e2) {
                int x = 4 * q + e2;                     // taps: cols x0+x-2 .. x0+x+2
                float ev = row[x + 2] + row[x + 6];
                float ov = row[x + 3] + row[x + 5];
                float mv = row[x + 4];
                h1[e2] = g1_0 * ev + g1_1 * ov + g1_2 * mv;
                h2[e2] = g2_0 * ev + g2_1 * ov + g2_2 * mv;
            }
            *(v4f*)(Hb + lane * HSTR + 4 * q)      = h1;
            *(v4f*)(Hb + HS + lane * HSTR + 4 * q) = h2;
        }
    }
    __syncthreads();   // covers atab fill + own-wave H visibility

    // ---- column pass: two 16x16 tiles, 10 chained WMMAs each; A reused across tiles ----
    const int hx   = lane & 15;
    const int koff = (lane < 16) ? 0 : 2;
    v2f a1[5], a2[5];
#pragma unroll
    for (int c = 0; c < 5; ++c) {
        a1[c] = atab[c * 32 + lane];
        a2[c] = atab[160 + c * 32 + lane];
    }

    const int radd = (lane < 16) ? 0 : 8;
#pragma unroll
    for (int t = 0; t < 2; ++t) {
        const float* Hbt = Hb + t * 16 + hx;
        v8f acc = {};
#pragma unroll
        for (int c = 0; c < 5; ++c) {
            const int k0 = 4 * c + koff;
            v2f b1, b2;
            b1.x = Hbt[k0 * HSTR];      b1.y = Hbt[(k0 + 1) * HSTR];
            b2.x = Hbt[HS + k0 * HSTR]; b2.y = Hbt[HS + (k0 + 1) * HSTR];
            acc = __builtin_amdgcn_wmma_f32_16x16x4_f32(false, a1[c], false, b1,
                                                        (short)0, acc, false, false);
            acc = __builtin_amdgcn_wmma_f32_16x16x4_f32(false, a2[c], false, b2,
                                                        (short)0, acc, false, false);
        }
        // store per 16x16 f32 C/D layout (VGPR j: M = j / j+8 by lane half)
        float* op = out + plane_base + (size_t)y0 * IMG_W + x0 + t * 16 + hx;
#pragma unroll
        for (int j = 0; j < 8; ++j) {
            __builtin_nontemporal_store(acc[j], op + (size_t)(j + radd) * IMG_W);
        }
    }
}

extern "C" void kernel_launch(void* const* d_in, const int* in_sizes, int n_in,
                              void* d_out, int out_size, void* d_ws, size_t ws_size,
                              hipStream_t stream) {
    const float* x = (const float*)d_in[0];
    float* y = (float*)d_out;
    // 96 planes * (32 x 16) strips = 49152 waves; 8 waves per 256-thread block
    const int blocks = (PLANES * 32 * 16) / WPB;   // 6144
    dog_wmma_kernel<<<blocks, 256, 0, stream>>>(x, y);
}